// ArmaNet_bench_1769526526164
// MI455X (gfx1250) — compile-verified
//
#include <hip/hip_runtime.h>

#define N_NODES 100000
#define N_EDGES 3200000
#define KS 3
#define HC 16
#define TL 4

typedef float v2f __attribute__((ext_vector_type(2)));
typedef float v8f __attribute__((ext_vector_type(8)));

// ---------------- utility kernels ----------------
__global__ void k_fill(float* __restrict__ p, float v, int n) {
    int i = blockIdx.x * blockDim.x + threadIdx.x;
    if (i < n) p[i] = v;
}

__global__ void k_copy(float* __restrict__ dst, const float* __restrict__ src, int n) {
    int i = blockIdx.x * blockDim.x + threadIdx.x;
    if (i < n) dst[i] = src[i];
}

__global__ void k_relu(float* __restrict__ p, int n) {
    int i = blockIdx.x * blockDim.x + threadIdx.x;
    if (i < n) p[i] = fmaxf(p[i], 0.0f);
}

// ---------------- gcn_norm ----------------
__global__ void k_degree(const int* __restrict__ col, const float* __restrict__ ew,
                         float* __restrict__ deg) {
    int e = blockIdx.x * blockDim.x + threadIdx.x;
    if (e < N_EDGES) atomicAdd(&deg[col[e]], ew[e]);
}

__global__ void k_dis(float* __restrict__ deg) {
    int n = blockIdx.x * blockDim.x + threadIdx.x;
    if (n < N_NODES) {
        float d = deg[n];
        deg[n] = (d > 0.0f) ? rsqrtf(fmaxf(d, 1e-12f)) : 0.0f;  // in-place -> dis
    }
}

__global__ void k_norm(const int* __restrict__ row, const int* __restrict__ col,
                       const float* __restrict__ ew, const float* __restrict__ dis,
                       float* __restrict__ norm) {
    int e = blockIdx.x * blockDim.x + threadIdx.x;
    if (e < N_EDGES) norm[e] = dis[row[e]] * ew[e] * dis[col[e]];
}

// ---------------- conv1 (F=16) ----------------
// out[k,n,f] = x[n]*init_w[k,0,f] ; root[k,n,f] = x[n]*root_w[k,0,f] + bias[k,f]
__global__ void k_conv1_init(const float* __restrict__ x,
                             const float* __restrict__ iw, const float* __restrict__ rw,
                             const float* __restrict__ bias,
                             float* __restrict__ out, float* __restrict__ root) {
    int i = blockIdx.x * blockDim.x + threadIdx.x;   // over N*16
    if (i >= N_NODES * HC) return;
    int n = i >> 4, f = i & 15;
    float xv = x[n];
#pragma unroll
    for (int k = 0; k < KS; ++k) {
        int o = (k * N_NODES + n) * HC + f;
        out[o]  = xv * iw[k * HC + f];
        root[o] = xv * rw[k * HC + f] + bias[k * HC + f];
    }
}

// In-place per-stack [16x16] @ [16x16] via V_WMMA_F32_16X16X4_F32 (fp32, no precision loss).
// One wave owns one 16-row tile of one stack; 4 chained k=4 WMMAs.
__global__ void k_arma_matmul16(float* __restrict__ buf, const float* __restrict__ w,
                                int tilesPerK) {
    int gwave = (blockIdx.x * blockDim.x + threadIdx.x) >> 5;  // wave-uniform
    int lane  = threadIdx.x & 31;
    if (gwave >= KS * tilesPerK) return;                        // whole-wave exit, EXEC stays all-1s
    int k    = gwave / tilesPerK;
    int tile = gwave - k * tilesPerK;
    const float* __restrict__ wk = w + k * (HC * HC);
    float* __restrict__ tp = buf + ((size_t)k * N_NODES + (size_t)tile * 16) * HC;

    int half = lane >> 4;    // 0: K-pair {0,1}/rows 0..7 ; 1: K-pair {2,3}/rows 8..15
    int ln   = lane & 15;    // M index for A, N index for B/C/D

    v8f c = {0.f, 0.f, 0.f, 0.f, 0.f, 0.f, 0.f, 0.f};
#pragma unroll
    for (int j = 0; j < 4; ++j) {
        int kk = 4 * j + 2 * half;
        v2f a, b;
        const float* ap = tp + ln * HC + kk;     // A: row ln, cols kk..kk+1 (8B aligned)
        a.x = ap[0];
        a.y = ap[1];
        b.x = wk[kk * HC + ln];                  // B: row kk,   col ln
        b.y = wk[(kk + 1) * HC + ln];            // B: row kk+1, col ln
        c = __builtin_amdgcn_wmma_f32_16x16x4_f32(
                false, a, false, b, (short)0, c, false, false);
    }
    // D layout: VGPR r holds rows r (lanes 0-15) / r+8 (lanes 16-31), col = ln
#pragma unroll
    for (int r = 0; r < 8; ++r) {
        tp[(r + 8 * half) * HC + ln] = c[r];
    }
}

// one thread per (edge, channel); loop over stacks
__global__ void k_scatter16(const int* __restrict__ row, const int* __restrict__ col,
                            const float* __restrict__ norm,
                            const float* __restrict__ src, float* __restrict__ dst) {
    int i = blockIdx.x * blockDim.x + threadIdx.x;
    if (i >= N_EDGES * HC) return;
    int e = i >> 4, f = i & 15;
    int r = row[e], c = col[e];
    float nm = norm[e];
#pragma unroll
    for (int k = 0; k < KS; ++k) {
        float v = src[(k * N_NODES + r) * HC + f] * nm;
        atomicAdd(&dst[(k * N_NODES + c) * HC + f], v);
    }
}

// mean over stacks + BatchNorm(eval) + ReLU
__global__ void k_bn_relu(const float* __restrict__ a,
                          const float* __restrict__ g, const float* __restrict__ b,
                          const float* __restrict__ m, const float* __restrict__ v,
                          float* __restrict__ h) {
    int i = blockIdx.x * blockDim.x + threadIdx.x;   // over N*16
    if (i >= N_NODES * HC) return;
    int f = i & 15;
    float s = (a[i] + a[N_NODES * HC + i] + a[2 * N_NODES * HC + i]) * (1.0f / 3.0f);
    float y = (s - m[f]) * rsqrtf(v[f] + 1e-5f) * g[f] + b[f];
    h[i] = fmaxf(y, 0.0f);
}

// ---------------- conv2 (F=1) ----------------
__global__ void k_conv2_init(const float* __restrict__ h,
                             const float* __restrict__ iw, const float* __restrict__ rw,
                             const float* __restrict__ bias,
                             float* __restrict__ out, float* __restrict__ root) {
    int i = blockIdx.x * blockDim.x + threadIdx.x;   // over K*N
    if (i >= KS * N_NODES) return;
    int k = i / N_NODES, n = i - k * N_NODES;
    float so = 0.0f, sr = 0.0f;
#pragma unroll
    for (int f = 0; f < HC; ++f) {
        float hv = h[n * HC + f];
        so += hv * iw[k * HC + f];
        sr += hv * rw[k * HC + f];
    }
    out[i]  = so;
    root[i] = sr + bias[k];
}

// per-stack scalar weight fused into the gather (useW=0 on first iteration)
__global__ void k_scatter1(const int* __restrict__ row, const int* __restrict__ col,
                           const float* __restrict__ norm,
                           const float* __restrict__ src, float* __restrict__ dst,
                           const float* __restrict__ w2, int useW) {
    int e = blockIdx.x * blockDim.x + threadIdx.x;
    if (e >= N_EDGES) return;
    int r = row[e], c = col[e];
    float nm = norm[e];
#pragma unroll
    for (int k = 0; k < KS; ++k) {
        float s = useW ? w2[k] : 1.0f;
        atomicAdd(&dst[k * N_NODES + c], src[k * N_NODES + r] * s * nm);
    }
}

__global__ void k_sigmoid(const float* __restrict__ a, float* __restrict__ y) {
    int n = blockIdx.x * blockDim.x + threadIdx.x;
    if (n >= N_NODES) return;
    float s = (a[n] + a[N_NODES + n] + a[2 * N_NODES + n]) * (1.0f / 3.0f);
    y[n] = 1.0f / (1.0f + expf(-s));
}

// ---------------- host orchestration ----------------
extern "C" void kernel_launch(void* const* d_in, const int* in_sizes, int n_in,
                              void* d_out, int out_size, void* d_ws, size_t ws_size,
                              hipStream_t stream) {
    (void)in_sizes; (void)n_in; (void)out_size; (void)ws_size;
    const float* x     = (const float*)d_in[0];
    const int*   ei    = (const int*)d_in[1];
    const float* ew    = (const float*)d_in[2];
    const float* c1_iw = (const float*)d_in[3];
    const float* c1_w  = (const float*)d_in[4];
    const float* c1_rw = (const float*)d_in[5];
    const float* c1_b  = (const float*)d_in[6];
    const float* bn_g  = (const float*)d_in[7];
    const float* bn_b  = (const float*)d_in[8];
    const float* bn_m  = (const float*)d_in[9];
    const float* bn_v  = (const float*)d_in[10];
    const float* c2_iw = (const float*)d_in[11];
    const float* c2_w  = (const float*)d_in[12];
    const float* c2_rw = (const float*)d_in[13];
    const float* c2_b  = (const float*)d_in[14];
    const int* row = ei;
    const int* col = ei + N_EDGES;

    // workspace carve-out (fp32, 256B-aligned regions), ~81 MB total
    float* ws = (float*)d_ws;
    size_t off = 0;
    auto carve = [&](size_t n) { float* p = ws + off; off += (n + 63) & ~(size_t)63; return p; };
    float* dis   = carve(N_NODES);                     // deg -> dis (in place)
    float* norm  = carve(N_EDGES);
    float* root1 = carve((size_t)KS * N_NODES * HC);
    float* bufA  = carve((size_t)KS * N_NODES * HC);
    float* bufB  = carve((size_t)KS * N_NODES * HC);
    float* h     = carve((size_t)N_NODES * HC);
    float* root2 = carve((size_t)KS * N_NODES);
    float* b2A   = carve((size_t)KS * N_NODES);
    float* b2B   = carve((size_t)KS * N_NODES);

    const int BT = 256;
    dim3 blk(BT);
    auto G = [&](long n) { return dim3((unsigned)((n + BT - 1) / BT)); };

    // gcn_norm
    k_fill<<<G(N_NODES), blk, 0, stream>>>(dis, 0.0f, N_NODES);
    k_degree<<<G(N_EDGES), blk, 0, stream>>>(col, ew, dis);
    k_dis<<<G(N_NODES), blk, 0, stream>>>(dis);
    k_norm<<<G(N_EDGES), blk, 0, stream>>>(row, col, ew, dis, norm);

    // conv1 (ARMA, act=ReLU)
    k_conv1_init<<<G((long)N_NODES * HC), blk, 0, stream>>>(x, c1_iw, c1_rw, c1_b, bufA, root1);
    float* cur = bufA;
    float* nxt = bufB;
    const int tilesPerK = N_NODES / 16;                 // 6250 exact
    const long wmmaThreads = (long)KS * tilesPerK * 32; // one wave per 16x16 tile
    for (int t = 0; t < TL; ++t) {
        if (t > 0)
            k_arma_matmul16<<<G(wmmaThreads), blk, 0, stream>>>(cur, c1_w, tilesPerK);
        k_copy<<<G((long)KS * N_NODES * HC), blk, 0, stream>>>(nxt, root1, KS * N_NODES * HC);
        k_scatter16<<<G((long)N_EDGES * HC), blk, 0, stream>>>(row, col, norm, cur, nxt);
        k_relu<<<G((long)KS * N_NODES * HC), blk, 0, stream>>>(nxt, KS * N_NODES * HC);
        float* tp = cur; cur = nxt; nxt = tp;
    }

    // mean over stacks + BN + ReLU
    k_bn_relu<<<G((long)N_NODES * HC), blk, 0, stream>>>(cur, bn_g, bn_b, bn_m, bn_v, h);

    // conv2 (ARMA, act=None, F=1)
    k_conv2_init<<<G((long)KS * N_NODES), blk, 0, stream>>>(h, c2_iw, c2_rw, c2_b, b2A, root2);
    float* c2 = b2A;
    float* n2 = b2B;
    for (int t = 0; t < TL; ++t) {
        k_copy<<<G((long)KS * N_NODES), blk, 0, stream>>>(n2, root2, KS * N_NODES);
        k_scatter1<<<G(N_EDGES), blk, 0, stream>>>(row, col, norm, c2, n2, c2_w, t > 0 ? 1 : 0);
        float* tp = c2; c2 = n2; n2 = tp;
    }

    // mean over stacks + sigmoid
    k_sigmoid<<<G(N_NODES), blk, 0, stream>>>(c2, (float*)d_out);
}